// FeaturePropagation_7335804142069
// MI455X (gfx1250) — compile-verified
//
#include <hip/hip_runtime.h>
#include <hip/hip_bf16.h>

typedef __attribute__((ext_vector_type(2))) float v2f;
typedef __attribute__((ext_vector_type(8))) float v8f;

#define NFINE   16384
#define MCOARSE 4096
#define CF      256
#define CC      512
#define IN1     768   // CC + CF
#define H1      512
#define H2      512

#define WS_STRIDE 36            // padded W-tile row stride (floats), 144B = 16B multiple
#define XS_STRIDE 288           // padded X-tile kpair row stride (floats), 1152B
#define WS_ELEMS  (64 * WS_STRIDE)
#define XS_ELEMS  (16 * XS_STRIDE)

__device__ __forceinline__ unsigned lds_off_u32(const void* p) {
    // low 32 bits of a generic pointer to __shared__ = wave-relative LDS offset
    return (unsigned)(unsigned long long)p;
}

// ---------------------------------------------------------------------------
// Kernel 1: brute-force kNN (k=3) fine->coarse, coarse xyz staged in LDS.
// ---------------------------------------------------------------------------
__global__ __launch_bounds__(256) void knn_kernel(
    const float* __restrict__ xyzf,   // [3, NFINE]
    const float* __restrict__ xyzc,   // [3, MCOARSE]
    int* __restrict__ idx, float* __restrict__ w)
{
    __shared__ float sc[3 * MCOARSE];          // 48 KB
    const int t = threadIdx.x;
    for (int i = t; i < 3 * MCOARSE; i += 256) sc[i] = xyzc[i];
    __syncthreads();

    const int n = blockIdx.x * 256 + t;
    const float px = xyzf[n];
    const float py = xyzf[NFINE + n];
    const float pz = xyzf[2 * NFINE + n];

    float d0 = 3.4e38f, d1 = 3.4e38f, d2 = 3.4e38f;
    int   i0 = 0, i1 = 0, i2 = 0;
    for (int m = 0; m < MCOARSE; ++m) {
        float dx = px - sc[m];
        float dy = py - sc[MCOARSE + m];
        float dz = pz - sc[2 * MCOARSE + m];
        float d = dx * dx + dy * dy + dz * dz;
        if (d < d0)      { d2 = d1; i2 = i1; d1 = d0; i1 = i0; d0 = d; i0 = m; }
        else if (d < d1) { d2 = d1; i2 = i1; d1 = d;  i1 = m; }
        else if (d < d2) { d2 = d;  i2 = m; }
    }
    float r0 = fmaxf(sqrtf(d0), 1e-8f);
    float r1 = fmaxf(sqrtf(d1), 1e-8f);
    float r2 = fmaxf(sqrtf(d2), 1e-8f);
    float w0 = 1.f / r0, w1 = 1.f / r1, w2 = 1.f / r2;
    float inv = 1.f / (w0 + w1 + w2);
    idx[n * 3 + 0] = i0; idx[n * 3 + 1] = i1; idx[n * 3 + 2] = i2;
    w[n * 3 + 0] = w0 * inv; w[n * 3 + 1] = w1 * inv; w[n * 3 + 2] = w2 * inv;
}

// ---------------------------------------------------------------------------
// Kernel 2: build X in K-pair-packed layout:
//   Xp[(c>>1)*2N + n*2 + (c&1)] = concat(interp, feats_fine)[c, n]
// ---------------------------------------------------------------------------
__global__ __launch_bounds__(256) void build_x_kernel(
    const float* __restrict__ fc,     // [CC, MCOARSE]
    const float* __restrict__ ff,     // [CF, NFINE]
    const int* __restrict__ idx, const float* __restrict__ w,
    float* __restrict__ Xp)           // packed [IN1/2][NFINE*2]
{
    const int c = blockIdx.y;
    const int n = blockIdx.x * 256 + threadIdx.x;
    float v;
    if (c < CC) {
        const int base = n * 3;
        const int i0 = idx[base], i1 = idx[base + 1], i2 = idx[base + 2];
        const float w0 = w[base], w1 = w[base + 1], w2 = w[base + 2];
        const float* row = fc + (size_t)c * MCOARSE;
        v = w0 * row[i0] + w1 * row[i1] + w2 * row[i2];
    } else {
        v = ff[(size_t)(c - CC) * NFINE + n];
    }
    Xp[(size_t)(c >> 1) * (2 * NFINE) + (size_t)n * 2 + (c & 1)] = v;
}

// ---------------------------------------------------------------------------
// Async staging: one K-chunk (32 k) of W (64x32) and packed-X (16 kpairs x 256)
// into LDS via gfx1250 global_load_async_to_lds_b128 (6 async ops per wave).
// ---------------------------------------------------------------------------
__device__ __forceinline__ void stage_async_chunk(
    const float* __restrict__ Wg, int Ci,
    const float* __restrict__ Xp, int Nn2,
    int bm, int bn, int k0,
    float* WsB, float* XsB, int t)
{
    // ---- W tile: 64 rows x 32 k, row-major, global row stride Ci ----
    {
        const unsigned long long base =
            (unsigned long long)(const void*)(Wg + (size_t)bm * 64 * Ci + k0);
#pragma unroll
        for (int p = 0; p < 2; ++p) {
            const int idx = t + p * 256;            // 0..511
            const int r   = idx >> 3;               // 0..63
            const int c16 = idx & 7;                // 0..7 (16B units)
            const unsigned voff = (unsigned)((r * Ci + c16 * 4) * 4);
            const unsigned ldso = lds_off_u32(&WsB[r * WS_STRIDE + c16 * 4]);
            asm volatile("global_load_async_to_lds_b128 %0, %1, %2"
                         :: "v"(ldso), "v"(voff), "s"(base) : "memory");
        }
    }
    // ---- X tile: 16 kpair rows x 256 floats, global row stride Nn2 ----
    {
        const unsigned long long base =
            (unsigned long long)(const void*)(Xp + (size_t)(k0 >> 1) * Nn2 + bn * 256);
#pragma unroll
        for (int p = 0; p < 4; ++p) {
            const int idx = t + p * 256;            // 0..1023
            const int r   = idx >> 6;               // 0..15
            const int c16 = idx & 63;               // 0..63 (16B units)
            const unsigned voff = (unsigned)((r * Nn2 + c16 * 4) * 4);
            const unsigned ldso = lds_off_u32(&XsB[r * XS_STRIDE + c16 * 4]);
            asm volatile("global_load_async_to_lds_b128 %0, %1, %2"
                         :: "v"(ldso), "v"(voff), "s"(base) : "memory");
        }
    }
}

// ---------------------------------------------------------------------------
// Kernel 3: fp32 WMMA GEMM with async double-buffered LDS staging.
//   Y[Co,N] = Wm[Co,Ci] @ X[Ci,N]
// X is K-pair packed; output optionally M-pair packed (for next layer).
// Block: 256 thr = 8 waves; tile 64(M) x 128(N); wave 16(M) x 64(N).
// ---------------------------------------------------------------------------
__global__ __launch_bounds__(256) void gemm_wmma_kernel(
    const float* __restrict__ Wm, const float* __restrict__ Xp,
    float* __restrict__ Y, int Ci, int Nn, int pack_out)
{
    __shared__ float Ws[2][WS_ELEMS];   // 2 x 9216 B
    __shared__ float Xs[2][XS_ELEMS];   // 2 x 18432 B

    const int t    = threadIdx.x;
    const int lane = t & 31;
    const int wid  = t >> 5;
    const int wm   = wid & 3;       // wave M index (0..3)
    const int wn   = wid >> 2;      // wave N index (0..1)
    const int half = lane >> 4;     // 0: K{0,1}, 1: K{2,3}
    const int lm   = lane & 15;

    const int bm = blockIdx.y;
    const int bn = blockIdx.x;
    const int Nn2 = Nn * 2;

    v8f acc[4];
#pragma unroll
    for (int s = 0; s < 4; ++s) acc[s] = (v8f){0.f,0.f,0.f,0.f,0.f,0.f,0.f,0.f};

    const int nch = Ci >> 5;        // K chunks of 32

    stage_async_chunk(Wm, Ci, Xp, Nn2, bm, bn, 0, Ws[0], Xs[0], t);

    for (int ch = 0; ch < nch; ++ch) {
        const int cur = ch & 1;
        if (ch + 1 < nch) {
            stage_async_chunk(Wm, Ci, Xp, Nn2, bm, bn, (ch + 1) * 32,
                              Ws[1 - cur], Xs[1 - cur], t);
            asm volatile("s_wait_asynccnt 0x6" ::: "memory"); // prev chunk done
        } else {
            asm volatile("s_wait_asynccnt 0x0" ::: "memory");
        }
        __syncthreads();

        const float* WsC = Ws[cur];
        const float* XsC = Xs[cur];
#pragma unroll
        for (int kk = 0; kk < 32; kk += 4) {
            const int kp = kk >> 1;
            v2f a = *(const v2f*)&WsC[(wm * 16 + lm) * WS_STRIDE + kk + half * 2];
#pragma unroll
            for (int s = 0; s < 4; ++s) {
                const int nl = wn * 64 + s * 16 + lm;
                v2f b = *(const v2f*)&XsC[(kp + half) * XS_STRIDE + nl * 2];
                acc[s] = __builtin_amdgcn_wmma_f32_16x16x4_f32(
                    false, a, false, b, (short)0, acc[s], false, false);
            }
        }
        __syncthreads();   // all waves done with this buffer before re-fill
    }

    // store C/D: VGPR r -> M = r + 8*half ; lane&15 -> N
    if (pack_out) {
#pragma unroll
        for (int s = 0; s < 4; ++s) {
            const int n = bn * 128 + wn * 64 + s * 16 + lm;
#pragma unroll
            for (int r = 0; r < 8; r += 2) {
                const int m = bm * 64 + wm * 16 + half * 8 + r;   // even
                v2f v; v.x = acc[s][r]; v.y = acc[s][r + 1];
                *(v2f*)&Y[(size_t)(m >> 1) * Nn2 + (size_t)n * 2] = v;
            }
        }
    } else {
#pragma unroll
        for (int s = 0; s < 4; ++s) {
            const int n = bn * 128 + wn * 64 + s * 16 + lm;
#pragma unroll
            for (int r = 0; r < 8; ++r) {
                const int m = bm * 64 + wm * 16 + half * 8 + r;
                Y[(size_t)m * Nn + n] = acc[s][r];
            }
        }
    }
}

// ---------------------------------------------------------------------------
// Kernel 4: per-channel BN stats (biased var) -> scale/shift
// ---------------------------------------------------------------------------
__global__ __launch_bounds__(256) void bn_stats_kernel(
    const float* __restrict__ Y, const float* __restrict__ g,
    const float* __restrict__ b, float* __restrict__ scale,
    float* __restrict__ shift, int Nn, int packed)
{
    __shared__ float ss[256];
    __shared__ float sq[256];
    const int c = blockIdx.x;
    const int t = threadIdx.x;
    const float* row = packed ? (Y + (size_t)(c >> 1) * (2 * (size_t)Nn) + (c & 1))
                              : (Y + (size_t)c * Nn);
    const int stride = packed ? 2 : 1;
    float s = 0.f, q = 0.f;
    for (int i = t; i < Nn; i += 256) {
        float v = row[(size_t)i * stride];
        s += v; q += v * v;
    }
    ss[t] = s; sq[t] = q;
    __syncthreads();
    for (int o = 128; o > 0; o >>= 1) {
        if (t < o) { ss[t] += ss[t + o]; sq[t] += sq[t + o]; }
        __syncthreads();
    }
    if (t == 0) {
        float mean = ss[0] / (float)Nn;
        float var  = sq[0] / (float)Nn - mean * mean;
        float inv  = rsqrtf(var + 1e-5f);
        float sc   = g[c] * inv;
        scale[c] = sc;
        shift[c] = b[c] - mean * sc;
    }
}

// ---------------------------------------------------------------------------
// Kernel 5: in-place normalize + ReLU (layout-aware channel index)
// ---------------------------------------------------------------------------
__global__ __launch_bounds__(256) void bn_relu_kernel(
    float* __restrict__ Y, const float* __restrict__ scale,
    const float* __restrict__ shift, int Nn, int packed)
{
    const size_t i = (size_t)blockIdx.x * 256 + threadIdx.x;
    const int c = packed ? (int)((i / (size_t)(2 * Nn)) * 2 + (i & 1))
                         : (int)(i / (size_t)Nn);
    Y[i] = fmaxf(fmaf(Y[i], scale[c], shift[c]), 0.f);
}

// ---------------------------------------------------------------------------
extern "C" void kernel_launch(void* const* d_in, const int* in_sizes, int n_in,
                              void* d_out, int out_size, void* d_ws, size_t ws_size,
                              hipStream_t stream)
{
    const float* xyzf = (const float*)d_in[0];
    const float* xyzc = (const float*)d_in[1];
    const float* ff   = (const float*)d_in[2];
    const float* fc   = (const float*)d_in[3];
    const float* W1   = (const float*)d_in[4];
    const float* g1   = (const float*)d_in[5];
    const float* b1   = (const float*)d_in[6];
    const float* W2   = (const float*)d_in[7];
    const float* g2   = (const float*)d_in[8];
    const float* b2   = (const float*)d_in[9];
    float* out = (float*)d_out;

    auto align_up = [](size_t x) { return (x + 255) & ~(size_t)255; };
    char* p = (char*)d_ws;
    int*   idxbuf = (int*)p;    p += align_up((size_t)NFINE * 3 * sizeof(int));
    float* wbuf   = (float*)p;  p += align_up((size_t)NFINE * 3 * sizeof(float));
    float* Xbuf   = (float*)p;  p += align_up((size_t)IN1 * NFINE * sizeof(float));
    float* Y1     = (float*)p;  p += align_up((size_t)H1 * NFINE * sizeof(float));
    float* scale1 = (float*)p;  p += align_up((size_t)H1 * sizeof(float));
    float* shift1 = (float*)p;  p += align_up((size_t)H1 * sizeof(float));
    float* scale2 = (float*)p;  p += align_up((size_t)H2 * sizeof(float));
    float* shift2 = (float*)p;  p += align_up((size_t)H2 * sizeof(float));

    // 1) kNN + weights
    knn_kernel<<<NFINE / 256, 256, 0, stream>>>(xyzf, xyzc, idxbuf, wbuf);

    // 2) interpolate + concat -> packed X
    build_x_kernel<<<dim3(NFINE / 256, IN1), 256, 0, stream>>>(
        fc, ff, idxbuf, wbuf, Xbuf);

    // 3) layer 1: Y1(packed) = W1 @ X ; BN + ReLU in place
    gemm_wmma_kernel<<<dim3(NFINE / 128, H1 / 64), 256, 0, stream>>>(
        W1, Xbuf, Y1, IN1, NFINE, /*pack_out=*/1);
    bn_stats_kernel<<<H1, 256, 0, stream>>>(Y1, g1, b1, scale1, shift1, NFINE, 1);
    bn_relu_kernel<<<(H1 * NFINE) / 256, 256, 0, stream>>>(Y1, scale1, shift1, NFINE, 1);

    // 4) layer 2: out = W2 @ Y1 ; BN + ReLU in place (standard layout)
    gemm_wmma_kernel<<<dim3(NFINE / 128, H2 / 64), 256, 0, stream>>>(
        W2, Y1, out, H2 /*Ci*/, NFINE, /*pack_out=*/0);
    bn_stats_kernel<<<H2, 256, 0, stream>>>(out, g2, b2, scale2, shift2, NFINE, 0);
    bn_relu_kernel<<<(H2 * NFINE) / 256, 256, 0, stream>>>(out, scale2, shift2, NFINE, 0);
}